// LSTM_18210661335311
// MI455X (gfx1250) — compile-verified
//
#include <hip/hip_runtime.h>

// ---------------------------------------------------------------------------
// 2-layer LSTM (IN=2, H=8) + MLP head, B=32768, T=256, fp32 via
// V_WMMA_F32_16X16X4_F32. One wave32 processes 16 batch rows; gates are
// computed as 16x16 WMMA tiles (M=batch, N=gate), chained over K with the
// bias preloaded into the accumulator.
// ---------------------------------------------------------------------------

typedef float v2f __attribute__((ext_vector_type(2)));
typedef float v8f __attribute__((ext_vector_type(8)));

#define WMMA_F32(A, B, C) \
  __builtin_amdgcn_wmma_f32_16x16x4_f32(false, (A), false, (B), (short)0, (C), false, false)

#define K_SIG   (-1.4426950408889634f)   // -log2(e)      : sigmoid via exp2
#define K_TANH2 (-2.8853900817779268f)   // -2*log2(e)    : tanh via exp2

__device__ __forceinline__ float fexp2(float x) { return __builtin_amdgcn_exp2f(x); }
__device__ __forceinline__ float frcp (float x) { return __builtin_amdgcn_rcpf(x); }

// lane <-> lane^8 exchange (SWAPX8), group-of-32 ds_swizzle encoding:
// offset = xor<<10 | or<<5 | and  = (8<<10) | 0x1f = 0x201f
__device__ __forceinline__ float swap8(float x) {
  return __int_as_float(__builtin_amdgcn_ds_swizzle(__float_as_int(x), 0x201f));
}

// One LSTM cell update on a pair of activated gate tiles.
// Tile A raw gates: N<8 -> i, N>=8 -> f   (all sigmoid)
// Tile B raw gates: N<8 -> g (tanh), N>=8 -> o (sigmoid)
// Cell state cs[] lives in lanes with N>=8; h is produced in those lanes.
__device__ __forceinline__ void cell_update(const v8f& gA, const v8f& gB,
                                            float* cs, float* hv,
                                            float k1, float a1, float b1) {
#pragma unroll
  for (int v = 0; v < 8; ++v) {
    float s0 = frcp(1.f + fexp2(K_SIG * gA[v]));        // sigma(i) | sigma(f)
    float s1 = a1 * frcp(1.f + fexp2(k1 * gB[v])) + b1; // tanh(g)  | sigma(o)
    float ig = s0 * s1;                                  // lanes N<8 : i*g
    float fc = s0 * cs[v];                               // lanes N>=8: f*c
    float cn = fc + swap8(ig);                           // valid in N>=8 lanes
    cs[v] = cn;
    float th = 2.f * frcp(1.f + fexp2(K_TANH2 * cn)) - 1.f;  // tanh(c)
    hv[v] = s1 * th;                                     // N>=8: o * tanh(c)
  }
}

__global__ __launch_bounds__(256) void lstm2_wmma_kernel(
    const float* __restrict__ x,       // [32768, 256, 2]
    const float* __restrict__ w_ih0,   // [32, 2]
    const float* __restrict__ w_hh0,   // [32, 8]
    const float* __restrict__ b_ih0,   // [32]
    const float* __restrict__ b_hh0,   // [32]
    const float* __restrict__ w_ih1,   // [32, 8]
    const float* __restrict__ w_hh1,   // [32, 8]
    const float* __restrict__ b_ih1,   // [32]
    const float* __restrict__ b_hh1,   // [32]
    const float* __restrict__ w_mlp,   // [2, 8]
    const float* __restrict__ b_mlp,   // [2]
    float* __restrict__ out,           // [32768, 2]
    int T)
{
  // Per-wave h-transpose buffers: [wave][layer][16 rows * stride 10]
  __shared__ float hbuf[8][2][160];

  const int lane = threadIdx.x & 31;
  const int wav  = threadIdx.x >> 5;
  const int hl   = lane >> 4;   // lane half (selects K pair within a chunk)
  const int n16  = lane & 15;   // N index (gates) / M index (A rows)
  const int rowBase = blockIdx.x * 128 + wav * 16;

  // ---- Preload B matrices (weights) into VGPRs in WMMA B-layout:
  //      lane -> N = n16, VGPR v + half hl -> K = 4*chunk + 2*hl + v.
  v2f B0[2][3];  // layer0: K = [x0, x1, h0..h7, 0, 0]  (3 chunks of 4)
  v2f B1[2][4];  // layer1: K = [h0(t) 0..7, h1(t-1) 0..7] (4 chunks of 4)
  float bias0[2], bias1[2];
#pragma unroll
  for (int tl = 0; tl < 2; ++tl) {
    const int n = tl * 16 + n16;           // gate row of the weight matrices
    bias0[tl] = b_ih0[n] + b_hh0[n];
    bias1[tl] = b_ih1[n] + b_hh1[n];
#pragma unroll
    for (int c = 0; c < 3; ++c)
#pragma unroll
      for (int v = 0; v < 2; ++v) {
        const int k = 4 * c + 2 * hl + v;
        B0[tl][c][v] = (k < 2) ? w_ih0[n * 2 + k]
                     : (k < 10) ? w_hh0[n * 8 + (k - 2)] : 0.f;
      }
#pragma unroll
    for (int c = 0; c < 4; ++c)
#pragma unroll
      for (int v = 0; v < 2; ++v) {
        const int k = 4 * c + 2 * hl + v;
        B1[tl][c][v] = (k < 8) ? w_ih1[n * 8 + k] : w_hh1[n * 8 + (k - 8)];
      }
  }

  // Per-lane activation constants for the mixed tile [g | o].
  const float k1 = (n16 < 8) ? K_TANH2 : K_SIG;
  const float a1 = (n16 < 8) ? 2.f : 1.f;
  const float b1 = (n16 < 8) ? -1.f : 0.f;
  const float hsel = (hl == 0) ? 1.f : 0.f;   // value-select helpers (no EXEC ops)

  float* hb0 = &hbuf[wav][0][0];
  float* hb1 = &hbuf[wav][1][0];
  for (int i = lane; i < 320; i += 32) hb0[i] = 0.f;  // h(t=-1) = 0, both layers

  float c0s[8], c1s[8], hv[8];
#pragma unroll
  for (int v = 0; v < 8; ++v) { c0s[v] = 0.f; c1s[v] = 0.f; }

  const float* xrow = x + (size_t)(rowBase + n16) * 2 * (size_t)T;

  for (int t = 0; t < T; ++t) {
    // Unconditional global load of x (lanes 16-31 read the same addresses as
    // 0-15 -> coalesced broadcast); selected into the A fragment by value so
    // the compiler emits global_load_b64 + ds_load_b64 + v_cndmask instead of
    // an address-select + flat_load.
    v2f xv = *(const v2f*)(xrow + 2 * t);

    // ---- Layer 0 A-matrix chunks (lane = row m=n16, K pair = 2*hl + {0,1})
    const float* hr0 = hb0 + n16 * 10;
    v2f l00 = *(const v2f*)(hr0 + 0);            // h0_{0,1}   (used by hl=1)
    v2f l01 = *(const v2f*)(hr0 + 2 + 2 * hl);   // h0_{2,3}/{4,5}
    v2f l02 = *(const v2f*)(hr0 + 6);            // h0_{6,7}   (used by hl=0)

    v2f a00, a01 = l01, a02;
    a00[0] = hl ? l00[0] : xv[0];                // K0: x0   | K2: h0_0
    a00[1] = hl ? l00[1] : xv[1];                // K1: x1   | K3: h0_1
    a02[0] = l02[0] * hsel;                      // K8: h0_6 | K10: 0
    a02[1] = l02[1] * hsel;                      // K9: h0_7 | K11: 0

    v8f g0, g1;
#pragma unroll
    for (int i = 0; i < 8; ++i) { g0[i] = bias0[0]; g1[i] = bias0[1]; }
    g0 = WMMA_F32(a00, B0[0][0], g0);
    g1 = WMMA_F32(a00, B0[1][0], g1);
    g0 = WMMA_F32(a01, B0[0][1], g0);
    g1 = WMMA_F32(a01, B0[1][1], g1);
    g0 = WMMA_F32(a02, B0[0][2], g0);
    g1 = WMMA_F32(a02, B0[1][2], g1);

    cell_update(g0, g1, c0s, hv, k1, a1, b1);

    // Store h0(t) (held in lanes N>=8) to LDS in A-layout-friendly [m][j].
    if (n16 >= 8) {
      float* p = hb0 + 80 * hl + (n16 - 8);
#pragma unroll
      for (int v = 0; v < 8; ++v) p[v * 10] = hv[v];
    }

    // ---- Layer 1 A-matrix chunks: K = [h0(t) 0..7, h1(t-1) 0..7]
    const float* hr0b = hb0 + n16 * 10 + 2 * hl;
    const float* hr1b = hb1 + n16 * 10 + 2 * hl;
    v2f a10 = *(const v2f*)(hr0b + 0);   // K 0..3
    v2f a11 = *(const v2f*)(hr0b + 4);   // K 4..7
    v2f a12 = *(const v2f*)(hr1b + 0);   // K 8..11
    v2f a13 = *(const v2f*)(hr1b + 4);   // K 12..15

    v8f q0, q1;
#pragma unroll
    for (int i = 0; i < 8; ++i) { q0[i] = bias1[0]; q1[i] = bias1[1]; }
    q0 = WMMA_F32(a10, B1[0][0], q0);
    q1 = WMMA_F32(a10, B1[1][0], q1);
    q0 = WMMA_F32(a11, B1[0][1], q0);
    q1 = WMMA_F32(a11, B1[1][1], q1);
    q0 = WMMA_F32(a12, B1[0][2], q0);
    q1 = WMMA_F32(a12, B1[1][2], q1);
    q0 = WMMA_F32(a13, B1[0][3], q0);
    q1 = WMMA_F32(a13, B1[1][3], q1);

    cell_update(q0, q1, c1s, hv, k1, a1, b1);

    if (n16 >= 8) {
      float* p = hb1 + 80 * hl + (n16 - 8);
#pragma unroll
      for (int v = 0; v < 8; ++v) p[v * 10] = hv[v];
    }
  }

  // ---- MLP head: out[b] = h1_final @ w_mlp.T + b_mlp  (lanes 0..15, 1 row each)
  if (lane < 16) {
    const float* hp = hb1 + lane * 10;
    float o0 = b_mlp[0], o1 = b_mlp[1];
#pragma unroll
    for (int j = 0; j < 8; ++j) {
      const float h = hp[j];
      o0 += h * w_mlp[j];
      o1 += h * w_mlp[8 + j];
    }
    float* op = out + (size_t)(rowBase + lane) * 2;
    op[0] = o0;
    op[1] = o1;
  }
}

extern "C" void kernel_launch(void* const* d_in, const int* in_sizes, int n_in,
                              void* d_out, int out_size, void* d_ws, size_t ws_size,
                              hipStream_t stream) {
  const float* x     = (const float*)d_in[0];
  const float* w_ih0 = (const float*)d_in[1];
  const float* w_hh0 = (const float*)d_in[2];
  const float* b_ih0 = (const float*)d_in[3];
  const float* b_hh0 = (const float*)d_in[4];
  const float* w_ih1 = (const float*)d_in[5];
  const float* w_hh1 = (const float*)d_in[6];
  const float* b_ih1 = (const float*)d_in[7];
  const float* b_hh1 = (const float*)d_in[8];
  const float* w_mlp = (const float*)d_in[9];
  const float* b_mlp = (const float*)d_in[10];
  float* out = (float*)d_out;

  const int T = 256;                       // sequence length (reference fixed)
  const int B = in_sizes[0] / (T * 2);     // 32768 batch rows
  const int rowsPerBlock = 128;            // 8 waves * 16 rows
  const int grid = B / rowsPerBlock;       // 256 blocks

  lstm2_wmma_kernel<<<dim3(grid), dim3(256), 0, stream>>>(
      x, w_ih0, w_hh0, b_ih0, b_hh0, w_ih1, w_hh1, b_ih1, b_hh1,
      w_mlp, b_mlp, out, T);
}